// MultiPositionExtractor_14826227106355
// MI455X (gfx1250) — compile-verified
//
#include <hip/hip_runtime.h>
#include <hip/hip_bf16.h>
#include <stdint.h>

typedef float v2f __attribute__((ext_vector_type(2)));
typedef float v8f __attribute__((ext_vector_type(8)));

__device__ __forceinline__ v8f wmma_16x16x4_f32(v2f a, v2f b, v8f c) {
  // D = A(16x4) x B(4x16) + C, full f32.
  return __builtin_amdgcn_wmma_f32_16x16x4_f32(false, a, false, b, (short)0, c,
                                               false, false);
}

// ---------------------------------------------------------------- prep ------
__global__ void prep_kernel(const float* __restrict__ points,
                            const float* __restrict__ prop,
                            float* __restrict__ xyz, float* __restrict__ feats) {
  int i = blockIdx.x * blockDim.x + threadIdx.x;
  if (i >= 32768) return;
  float px = points[i * 3 + 0], py = points[i * 3 + 1], pz = points[i * 3 + 2];
  float x = prop[0], y = prop[1], w = prop[3];
  xyz[i * 3 + 0] = px - x;
  xyz[i * 3 + 1] = py - y;
  xyz[i * 3 + 2] = pz;
  float ox[4] = {x + 0.5f * w, x - 0.5f * w, x, x};
  float oy[4] = {y, y, y + 0.5f * w, y - 0.5f * w};
#pragma unroll
  for (int j = 0; j < 4; ++j) {
    feats[i * 12 + j * 3 + 0] = px - ox[j];
    feats[i * 12 + j * 3 + 1] = py - oy[j];
    feats[i * 12 + j * 3 + 2] = pz;
  }
}

// ---------------------------------------------------------------- FPS -------
// Single persistent workgroup; per-thread running min distances in registers;
// wave32 shfl argmax tree + one 32-partial LDS stage per iteration.
template <int N, int NPOINT>
__global__ void __launch_bounds__(1024) fps_kernel(const float* __restrict__ xyz,
                                                   int* __restrict__ outIdx,
                                                   float* __restrict__ newXyz) {
  constexpr int T = 1024;
  constexpr int PPT = N / T;
  const int tid = threadIdx.x;
  const int lane = tid & 31, wave = tid >> 5;
  float dists[PPT];
#pragma unroll
  for (int i = 0; i < PPT; ++i) dists[i] = 1e10f;
  __shared__ float s_val[32];
  __shared__ int s_idx[32];
  __shared__ float s_cent[3];
  __shared__ int s_far;
  if (tid == 0) {
    s_far = 0;
    s_cent[0] = xyz[0];
    s_cent[1] = xyz[1];
    s_cent[2] = xyz[2];
  }
  __syncthreads();
  for (int it = 0; it < NPOINT; ++it) {
    float cx = s_cent[0], cy = s_cent[1], cz = s_cent[2];
    if (tid == 0) {
      outIdx[it] = s_far;
      newXyz[it * 3 + 0] = cx;
      newXyz[it * 3 + 1] = cy;
      newXyz[it * 3 + 2] = cz;
    }
    float bv = -1.f;
    int bi = 0;
#pragma unroll
    for (int i = 0; i < PPT; ++i) {
      int p = tid + i * T;
      float dx = xyz[p * 3 + 0] - cx;
      float dy = xyz[p * 3 + 1] - cy;
      float dz = xyz[p * 3 + 2] - cz;
      float d = dx * dx + dy * dy + dz * dz;
      float dm = fminf(dists[i], d);
      dists[i] = dm;
      if (dm > bv) { bv = dm; bi = p; }
    }
#pragma unroll
    for (int o = 16; o > 0; o >>= 1) {
      float ov = __shfl_down(bv, o, 32);
      int oi = __shfl_down(bi, o, 32);
      if (ov > bv || (ov == bv && oi < bi)) { bv = ov; bi = oi; }
    }
    if (lane == 0) { s_val[wave] = bv; s_idx[wave] = bi; }
    __syncthreads();
    if (wave == 0) {
      float v = s_val[lane];
      int ix = s_idx[lane];
#pragma unroll
      for (int o = 16; o > 0; o >>= 1) {
        float ov = __shfl_down(v, o, 32);
        int oi = __shfl_down(ix, o, 32);
        if (ov > v || (ov == v && oi < ix)) { v = ov; ix = oi; }
      }
      if (lane == 0) {
        s_far = ix;
        s_cent[0] = xyz[ix * 3 + 0];
        s_cent[1] = xyz[ix * 3 + 1];
        s_cent[2] = xyz[ix * 3 + 2];
      }
    }
    __syncthreads();
  }
}

// ------------------------------------------------------------ ball query ----
__global__ void ball_query_kernel(const float* __restrict__ xyz, int N,
                                  const float* __restrict__ newXyz, float r2,
                                  int nsample, int* __restrict__ gidx) {
  const int c = blockIdx.x;
  const int lane = threadIdx.x;
  const float cx = newXyz[c * 3 + 0], cy = newXyz[c * 3 + 1],
              cz = newXyz[c * 3 + 2];
  int* row = gidx + (size_t)c * nsample;
  int count = 0;
  int firstLocal = 0x7fffffff;
  for (int base = 0; base < N && count < nsample; base += 32) {
    int p = base + lane;
    bool in = false;
    if (p < N) {
      float dx = xyz[p * 3 + 0] - cx;
      float dy = xyz[p * 3 + 1] - cy;
      float dz = xyz[p * 3 + 2] - cz;
      in = (dx * dx + dy * dy + dz * dz) <= r2;
    }
    unsigned m = (unsigned)__ballot(in);
    int pre = __popc(m & ((1u << lane) - 1u));
    if (in) {
      int slot = count + pre;
      if (slot < nsample) row[slot] = p;
      if (slot == 0) firstLocal = p;
    }
    count += __popc(m);
  }
#pragma unroll
  for (int o = 16; o > 0; o >>= 1) {
    int ov = __shfl_down(firstLocal, o, 32);
    firstLocal = firstLocal < ov ? firstLocal : ov;
  }
  int first = __shfl(firstLocal, 0, 32);
  if (first == 0x7fffffff) first = 0;
  int cnt = count < nsample ? count : nsample;
  for (int k = cnt + lane; k < nsample; k += 32) row[k] = first;
}

// ------------------------------------------- grouped features (padded rows) -
__global__ void buildH1_kernel(const float* __restrict__ xyz,
                               const float* __restrict__ feats,
                               const float* __restrict__ new1,
                               const int* __restrict__ gidx,
                               float* __restrict__ H) {  // stride 16
  int i = blockIdx.x * blockDim.x + threadIdx.x;  // 2048*32 rows
  if (i >= 2048 * 32) return;
  int s = i >> 5;
  int g = gidx[i];
  float* h = H + (size_t)i * 16;
  h[0] = xyz[g * 3 + 0] - new1[s * 3 + 0];
  h[1] = xyz[g * 3 + 1] - new1[s * 3 + 1];
  h[2] = xyz[g * 3 + 2] - new1[s * 3 + 2];
#pragma unroll
  for (int j = 0; j < 12; ++j) h[3 + j] = feats[g * 12 + j];
  h[15] = 0.f;
}

__global__ void buildH2_kernel(const float* __restrict__ xyz1,
                               const float* __restrict__ f1,
                               const float* __restrict__ new2,
                               const int* __restrict__ gidx,
                               float* __restrict__ H) {  // stride 132
  int i = blockIdx.x * blockDim.x + threadIdx.x;  // 512*64 rows
  if (i >= 512 * 64) return;
  int s = i >> 6;
  int g = gidx[i];
  float* h = H + (size_t)i * 132;
  h[0] = xyz1[g * 3 + 0] - new2[s * 3 + 0];
  h[1] = xyz1[g * 3 + 1] - new2[s * 3 + 1];
  h[2] = xyz1[g * 3 + 2] - new2[s * 3 + 2];
  for (int j = 0; j < 128; ++j) h[3 + j] = f1[(size_t)g * 128 + j];
  h[131] = 0.f;
}

__global__ void buildH3_kernel(const float* __restrict__ new2,
                               const float* __restrict__ f2,
                               float* __restrict__ H) {  // stride 260
  int i = blockIdx.x * blockDim.x + threadIdx.x;  // 512 rows
  if (i >= 512) return;
  float* h = H + (size_t)i * 260;
  h[0] = new2[i * 3 + 0];
  h[1] = new2[i * 3 + 1];
  h[2] = new2[i * 3 + 2];
  for (int j = 0; j < 256; ++j) h[3 + j] = f2[(size_t)i * 256 + j];
  h[259] = 0.f;
}

__global__ void zero_kernel(float* __restrict__ p, int n) {
  int i = blockIdx.x * blockDim.x + threadIdx.x;
  if (i < n) p[i] = 0.f;
}

// zero-pad weight rows: Wp (O x Cp) from W (O x C)
__global__ void packw_kernel(const float* __restrict__ W,
                             float* __restrict__ Wp, int O, int C, int Cp) {
  int i = blockIdx.x * blockDim.x + threadIdx.x;
  if (i >= O * Cp) return;
  int o = i / Cp, c = i - o * Cp;
  Wp[i] = (c < C) ? W[o * C + c] : 0.f;
}

// ----------------------------------------------------------- WMMA GEMM -----
// out(M,O) = relu(A(M,Cp) * W(O,Cp)^T + bias); A and W rows both have stride
// Cp (multiple of 4, zero-padded) => all loads are unconditional b64.
// One wave computes a 16x64 tile (A fragment reused across 4 N-tiles).
// If POOL, fold max over Kgroup consecutive rows via bit-monotonic atomicMax
// (values >= 0 after ReLU).
template <bool POOL>
__global__ void gemm_bias_relu(const float* __restrict__ A,
                               const float* __restrict__ W,
                               const float* __restrict__ bias,
                               float* __restrict__ out, int Cp, int O,
                               int Kgroup) {
  const int lane = threadIdx.x & 31;
  const int half = lane >> 4;  // 0: K0/K1 + rows 0..7 | 1: K2/K3 + rows 8..15
  const int mn = lane & 15;
  const int m0 = blockIdx.x << 4;
  const int o0 = blockIdx.y << 6;
  v8f acc[4] = {};
  const float* arow = A + (size_t)(m0 + mn) * Cp + 2 * half;
  const float* wr0 = W + (size_t)(o0 + 0 * 16 + mn) * Cp + 2 * half;
  const float* wr1 = W + (size_t)(o0 + 1 * 16 + mn) * Cp + 2 * half;
  const float* wr2 = W + (size_t)(o0 + 2 * 16 + mn) * Cp + 2 * half;
  const float* wr3 = W + (size_t)(o0 + 3 * 16 + mn) * Cp + 2 * half;
#pragma unroll 2
  for (int kb = 0; kb < Cp; kb += 4) {
    v2f a = *(const v2f*)(arow + kb);
    acc[0] = wmma_16x16x4_f32(a, *(const v2f*)(wr0 + kb), acc[0]);
    acc[1] = wmma_16x16x4_f32(a, *(const v2f*)(wr1 + kb), acc[1]);
    acc[2] = wmma_16x16x4_f32(a, *(const v2f*)(wr2 + kb), acc[2]);
    acc[3] = wmma_16x16x4_f32(a, *(const v2f*)(wr3 + kb), acc[3]);
  }
#pragma unroll
  for (int t = 0; t < 4; ++t) {
    const int col = o0 + t * 16 + mn;
    const float bv = bias[col];
#pragma unroll
    for (int i = 0; i < 8; ++i) {
      const int r = m0 + i + 8 * half;
      float v = acc[t][i] + bv;
      v = v > 0.f ? v : 0.f;
      if (POOL) {
        int s = r / Kgroup;
        atomicMax((unsigned int*)&out[(size_t)s * O + col], __float_as_uint(v));
      } else {
        out[(size_t)r * O + col] = v;
      }
    }
  }
}

// ---------------------------------------------------------------------------
extern "C" void kernel_launch(void* const* d_in, const int* in_sizes, int n_in,
                              void* d_out, int out_size, void* d_ws,
                              size_t ws_size, hipStream_t stream) {
  (void)in_sizes; (void)n_in; (void)out_size; (void)ws_size;
  const float* points = (const float*)d_in[0];
  const float* prop = (const float*)d_in[1];
  const float* w[3][3];
  const float* b[3][3];
  int di = 2;  // dict order: saN_w0, saN_b0, saN_w1, saN_b1, saN_w2, saN_b2
  for (int s = 0; s < 3; ++s)
    for (int l = 0; l < 3; ++l) {
      w[s][l] = (const float*)d_in[di++];
      b[s][l] = (const float*)d_in[di++];
    }

  char* ws = (char*)d_ws;
  size_t off = 0;
  auto alloc = [&](size_t bytes) -> void* {
    void* p = ws + off;
    off += (bytes + 255) & ~(size_t)255;
    return p;
  };
  float* xyz = (float*)alloc(32768 * 3 * 4);
  float* feats = (float*)alloc(32768 * 12 * 4);
  float* new1 = (float*)alloc(2048 * 3 * 4);
  int* idx1 = (int*)alloc(2048 * 4);
  int* gidx1 = (int*)alloc(2048 * 32 * 4);
  float* f1 = (float*)alloc((size_t)2048 * 128 * 4);
  float* new2 = (float*)alloc(512 * 3 * 4);
  int* idx2 = (int*)alloc(512 * 4);
  int* gidx2 = (int*)alloc(512 * 64 * 4);
  float* f2 = (float*)alloc((size_t)512 * 256 * 4);
  float* wp10 = (float*)alloc((size_t)64 * 16 * 4);     // sa1_w0 padded 15->16
  float* wp20 = (float*)alloc((size_t)128 * 132 * 4);   // sa2_w0 padded 131->132
  float* wp30 = (float*)alloc((size_t)256 * 260 * 4);   // sa3_w0 padded 259->260
  float* P0 = (float*)alloc((size_t)4325376 * 4);  // max(32768*132, 65536*64)
  float* P1 = (float*)alloc((size_t)4194304 * 4);  // max(65536*64, 32768*128)
  float* outF = (float*)d_out;

  // prep + weight packing (independent of downstream, same stream ordering)
  prep_kernel<<<128, 256, 0, stream>>>(points, prop, xyz, feats);
  packw_kernel<<<(64 * 16 + 255) / 256, 256, 0, stream>>>(w[0][0], wp10, 64, 15, 16);
  packw_kernel<<<(128 * 132 + 255) / 256, 256, 0, stream>>>(w[1][0], wp20, 128, 131, 132);
  packw_kernel<<<(256 * 260 + 255) / 256, 256, 0, stream>>>(w[2][0], wp30, 256, 259, 260);

  // ---- SA1 ----
  fps_kernel<32768, 2048><<<1, 1024, 0, stream>>>(xyz, idx1, new1);
  ball_query_kernel<<<2048, 32, 0, stream>>>(xyz, 32768, new1, 0.4f * 0.4f, 32,
                                             gidx1);
  buildH1_kernel<<<256, 256, 0, stream>>>(xyz, feats, new1, gidx1, P0);
  gemm_bias_relu<false><<<dim3(65536 / 16, 64 / 64), 32, 0, stream>>>(
      P0, wp10, b[0][0], P1, 16, 64, 0);
  gemm_bias_relu<false><<<dim3(65536 / 16, 64 / 64), 32, 0, stream>>>(
      P1, w[0][1], b[0][1], P0, 64, 64, 0);
  zero_kernel<<<(2048 * 128 + 255) / 256, 256, 0, stream>>>(f1, 2048 * 128);
  gemm_bias_relu<true><<<dim3(65536 / 16, 128 / 64), 32, 0, stream>>>(
      P0, w[0][2], b[0][2], f1, 64, 128, 32);

  // ---- SA2 ----
  fps_kernel<2048, 512><<<1, 1024, 0, stream>>>(new1, idx2, new2);
  ball_query_kernel<<<512, 32, 0, stream>>>(new1, 2048, new2, 0.8f * 0.8f, 64,
                                            gidx2);
  buildH2_kernel<<<128, 256, 0, stream>>>(new1, f1, new2, gidx2, P0);
  gemm_bias_relu<false><<<dim3(32768 / 16, 128 / 64), 32, 0, stream>>>(
      P0, wp20, b[1][0], P1, 132, 128, 0);
  gemm_bias_relu<false><<<dim3(32768 / 16, 128 / 64), 32, 0, stream>>>(
      P1, w[1][1], b[1][1], P0, 128, 128, 0);
  zero_kernel<<<(512 * 256 + 255) / 256, 256, 0, stream>>>(f2, 512 * 256);
  gemm_bias_relu<true><<<dim3(32768 / 16, 256 / 64), 32, 0, stream>>>(
      P0, w[1][2], b[1][2], f2, 128, 256, 64);

  // ---- SA3 (global) ----
  buildH3_kernel<<<2, 256, 0, stream>>>(new2, f2, P1);
  gemm_bias_relu<false><<<dim3(512 / 16, 256 / 64), 32, 0, stream>>>(
      P1, wp30, b[2][0], P0, 260, 256, 0);
  gemm_bias_relu<false><<<dim3(512 / 16, 512 / 64), 32, 0, stream>>>(
      P0, w[2][1], b[2][1], P1, 256, 512, 0);
  zero_kernel<<<(1024 + 255) / 256, 256, 0, stream>>>(outF, 1024);
  gemm_bias_relu<true><<<dim3(512 / 16, 1024 / 64), 32, 0, stream>>>(
      P1, w[2][2], b[2][2], outF, 512, 1024, 512);
}